// SolveRegistration_58368605553276
// MI455X (gfx1250) — compile-verified
//
#include <hip/hip_runtime.h>
#include <math.h>

typedef __attribute__((ext_vector_type(2))) float v2f;
typedef __attribute__((ext_vector_type(8))) float v8f;
typedef int sr_v4i __attribute__((vector_size(16)));
typedef __attribute__((address_space(1))) sr_v4i* sr_gptr;
typedef __attribute__((address_space(3))) sr_v4i* sr_lptr;

#define TP 256          // points per tile (per block iteration)
#define XTOL_C 1e-07f
#define MAXITER_C 5

// LDS layout (float indices); bases padded so p1/p0/ndf land in distinct
// bank windows (784%64=16, 1568%64=32, 1840%64=48).
#define OFF_P1   0       // 768 floats  (256 points x 3, AoS)
#define OFF_P0   784     // 768 floats
#define OFF_NDF  1568    // 256 floats
#define OFF_ZERO 1840    // 16 floats, zero word region for unused features
#define OFF_GRED 1856    // 512 floats, cross-wave Gram reduction
#define LDS_FLOATS 2368

#if __has_builtin(__builtin_amdgcn_global_load_async_to_lds_b128)
#define SR_HAS_ASYNC 1
#else
#define SR_HAS_ASYNC 0
#endif

__device__ __forceinline__ void sr_wait_async() {
#if __has_builtin(__builtin_amdgcn_s_wait_asynccnt)
    __builtin_amdgcn_s_wait_asynccnt(0);
#else
    asm volatile("s_wait_asynccnt 0" ::: "memory");
#endif
}

__device__ __forceinline__ void sr_async16(const float* g, float* l) {
#if SR_HAS_ASYNC
    __builtin_amdgcn_global_load_async_to_lds_b128(
        (sr_gptr)(float*)g, (sr_lptr)l, 0, 0);
#else
    (void)g; (void)l;
#endif
}

// ---------------------------------------------------------------------------
// Kernel 0: zero the per-batch 8x8 Gram accumulators in workspace
// ---------------------------------------------------------------------------
__global__ void sr_init_kernel(float* accum, int n) {
    int i = blockIdx.x * blockDim.x + threadIdx.x;
    if (i < n) accum[i] = 0.0f;
}

// ---------------------------------------------------------------------------
// Kernel 1: streaming Gram matrix G = sum_n u u^T via V_WMMA_F32_16X16X4_F32
// u = [p1x, p1y, p1z, ndf, 1, p0x, p0y, p0z, 0...]. One 16x16 f32 WMMA
// accumulator per wave; the same VGPR pair serves as A(16x4) and B(4x16).
// Full tiles are staged AoS into LDS with GLOBAL_LOAD_ASYNC_TO_LDS_B128.
// ---------------------------------------------------------------------------
__global__ void sr_gram_kernel(const float* __restrict__ p0,
                               const float* __restrict__ p1,
                               const float* __restrict__ ndf,
                               float* __restrict__ accum,
                               int N, int numTiles) {
    __shared__ float lds[LDS_FLOATS];

    const int b    = blockIdx.y;
    const int tid  = threadIdx.x;      // 0..255
    const int wave = tid >> 5;
    const int lane = tid & 31;

    const size_t base3 = (size_t)b * (size_t)N * 3;
    const float* p0b  = p0  + base3;
    const float* p1b  = p1  + base3;
    const float* ndfb = ndf + (size_t)b * (size_t)N;

    // zero the shared zero-word region once (read by unused-feature lanes)
    if (tid < 16) lds[OFF_ZERO + tid] = 0.0f;

    // Per-lane feature addressing for the WMMA A/B feed:
    //   A[m][k] layout: lanes 0-15 hold M=m for K={0,1}; lanes 16-31 for K={2,3}
    const int m   = lane & 15;
    const int klo = (lane >> 4) * 2;
    const int wbase = wave * 32;            // this wave's 32 points in the tile
    int fbase, fstride;
    if (m < 3)                { fbase = OFF_P1 + m;       fstride = 3; }
    else if (m == 3)          { fbase = OFF_NDF;          fstride = 1; }
    else if (m >= 5 && m < 8) { fbase = OFF_P0 + (m - 5); fstride = 3; }
    else                      { fbase = OFF_ZERO + m;     fstride = 0; } // m==4, m>=8
    const bool isOne = (m == 4);            // "count" feature, synthesized

    const bool canAsync = SR_HAS_ASYNC && ((N & 3) == 0);  // 16B-aligned tiles

    v8f c = {0.f, 0.f, 0.f, 0.f, 0.f, 0.f, 0.f, 0.f};

    for (int t = blockIdx.x; t < numTiles; t += gridDim.x) {
        const int tBase = t * TP;
        const bool fullTile = canAsync && (tBase + TP <= N);

        __syncthreads();   // protect previous tile's LDS reads

        if (fullTile) {
            // Async AoS staging: p1 3072B (192 x b128), p0 3072B, ndf 1024B.
            const float* gp1 = p1b  + (size_t)tBase * 3;
            const float* gp0 = p0b  + (size_t)tBase * 3;
            const float* gnd = ndfb + (size_t)tBase;
            if (tid < 192) {
                sr_async16(gp1 + tid * 4, &lds[OFF_P1 + tid * 4]);
                sr_async16(gp0 + tid * 4, &lds[OFF_P0 + tid * 4]);
            } else {
                const int u = tid - 192;
                sr_async16(gnd + u * 4, &lds[OFF_NDF + u * 4]);
            }
            sr_wait_async();
        } else {
            // Register path (tail tile / unaligned N): masked loads, zeros OOB
            const int pt = tBase + tid;
            float a0 = 0.f, a1 = 0.f, a2 = 0.f, nd = 0.f;
            float b0 = 0.f, b1 = 0.f, b2 = 0.f;
            if (pt < N) {
                const size_t o3 = (size_t)pt * 3;
                a0 = p1b[o3 + 0]; a1 = p1b[o3 + 1]; a2 = p1b[o3 + 2];
                nd = ndfb[pt];
                b0 = p0b[o3 + 0]; b1 = p0b[o3 + 1]; b2 = p0b[o3 + 2];
            }
            lds[OFF_P1 + tid * 3 + 0] = a0;
            lds[OFF_P1 + tid * 3 + 1] = a1;
            lds[OFF_P1 + tid * 3 + 2] = a2;
            lds[OFF_NDF + tid]        = nd;
            lds[OFF_P0 + tid * 3 + 0] = b0;
            lds[OFF_P0 + tid * 3 + 1] = b1;
            lds[OFF_P0 + tid * 3 + 2] = b2;
        }
        __syncthreads();

        // 8 WMMAs x 4 points = this wave's 32 points. No divergence here:
        // unused-feature lanes read the broadcast zero word; the "1" feature
        // is a branch-free select on point validity.
        #pragma unroll
        for (int j = 0; j < 8; ++j) {
            const int idx0 = wbase + j * 4 + klo;
            const int idx1 = idx0 + 1;
            float a0v = lds[fbase + idx0 * fstride];
            float a1v = lds[fbase + idx1 * fstride];
            a0v = isOne ? ((tBase + idx0) < N ? 1.0f : 0.0f) : a0v;
            a1v = isOne ? ((tBase + idx1) < N ? 1.0f : 0.0f) : a1v;
            v2f a = {a0v, a1v};
            c = __builtin_amdgcn_wmma_f32_16x16x4_f32(
                    false, a, false, a, (short)0, c, false, false);
        }
    }

    // C/D layout: VGPR r, lanes 0-15 hold row M=r, col N=lane.
    // Top-left 8x8 block lives in lanes 0..7, VGPRs 0..7.
    if (lane < 8) {
        #pragma unroll
        for (int r = 0; r < 8; ++r)
            lds[OFF_GRED + wave * 64 + r * 8 + lane] = c[r];
    }
    __syncthreads();
    if (tid < 64) {
        float s = 0.f;
        #pragma unroll
        for (int w = 0; w < 8; ++w) s += lds[OFF_GRED + w * 64 + tid];
        atomicAdd(&accum[(size_t)b * 64 + tid], s);
    }
}

// ---------------------------------------------------------------------------
// Kernel 2: finalize. One wave of 32 lanes; lane = batch index.
// Closed-form Gauss-Newton iterations (J is rank-1), wave-wide NaN-propagating
// max for the shared `done` flag, 6x6 Gauss-Jordan for inv(H).
// ---------------------------------------------------------------------------
__global__ void sr_finalize_kernel(const float* __restrict__ accum,
                                   const float* __restrict__ dt,
                                   float* __restrict__ out,
                                   int Bn, int N) {
    const int lane = threadIdx.x;
    const bool valid = lane < Bn;
    const int b = valid ? lane : 0;
    const float* G = accum + (size_t)b * 64;
    const float Nf = (float)N;

    float Gm[4][4];
    #pragma unroll
    for (int i = 0; i < 4; ++i)
        #pragma unroll
        for (int j = 0; j < 4; ++j)
            Gm[i][j] = G[i * 8 + j];

    float Sp1[3] = {G[0 * 8 + 4], G[1 * 8 + 4], G[2 * 8 + 4]};
    float Sndf   =  G[3 * 8 + 4];
    float Sp0[3] = {G[5 * 8 + 4], G[6 * 8 + 4], G[7 * 8 + 4]};

    float p1m[3], p0m[3];
    #pragma unroll
    for (int j = 0; j < 3; ++j) { p1m[j] = Sp1[j] / Nf; p0m[j] = Sp0[j] / Nf; }

    const float sm = (p1m[0] + p1m[1] + p1m[2]) / 3.0f;

    // a[n] = (p1x+p1y+p1z)/3 - ndf ;  dist0 = a - sm
    const float Sa  = (Sp1[0] + Sp1[1] + Sp1[2]) / 3.0f - Sndf;
    const float Sa2 = (Gm[0][0] + Gm[1][1] + Gm[2][2]
                       + 2.0f * (Gm[0][1] + Gm[0][2] + Gm[1][2])) / 9.0f
                      - (2.0f / 3.0f) * (Gm[3][0] + Gm[3][1] + Gm[3][2])
                      + Gm[3][3];
    float Sap1[3];
    #pragma unroll
    for (int j = 0; j < 3; ++j)
        Sap1[j] = (Gm[0][j] + Gm[1][j] + Gm[2][j]) / 3.0f - Gm[3][j];
    const float San = (Gm[3][0] + Gm[3][1] + Gm[3][2]) / 3.0f - Gm[3][3];

    const float D1 = Sa - Nf * sm;                        // sum dist0
    const float S2 = Sa2 - 2.0f * sm * Sa + Nf * sm * sm; // sum dist0^2
    float M[3];
    #pragma unroll
    for (int j = 0; j < 3; ++j) M[j] = Sap1[j] - p1m[j] * Sa; // sum dist0*q1_j
    const float DN = San - sm * Sndf;                     // sum dist0*ndf

    float invdt[6];
    #pragma unroll
    for (int l = 0; l < 6; ++l) invdt[l] = 1.0f / dt[l];

    // H = S2 * invdt invdt^T ; invert via Gauss-Jordan w/ partial pivoting
    float A[6][12];
    for (int i = 0; i < 6; ++i) {
        for (int j = 0; j < 6; ++j) A[i][j] = S2 * invdt[i] * invdt[j];
        for (int j = 0; j < 6; ++j) A[i][6 + j] = (i == j) ? 1.0f : 0.0f;
    }
    for (int col = 0; col < 6; ++col) {
        int piv = col; float mx = fabsf(A[col][col]);
        for (int r = col + 1; r < 6; ++r) {
            float v = fabsf(A[r][col]);
            if (v > mx) { mx = v; piv = r; }
        }
        if (piv != col)
            for (int j = 0; j < 12; ++j) {
                float tsw = A[col][j]; A[col][j] = A[piv][j]; A[piv][j] = tsw;
            }
        float inv = 1.0f / A[col][col];
        for (int j = 0; j < 12; ++j) A[col][j] *= inv;
        for (int r = 0; r < 6; ++r) {
            if (r == col) continue;
            float f = A[r][col];
            for (int j = 0; j < 12; ++j) A[r][j] -= f * A[col][j];
        }
    }
    float hk[6];
    for (int k = 0; k < 6; ++k) {
        float s = 0.f;
        for (int l = 0; l < 6; ++l) s += A[k][6 + l] * invdt[l];
        hk[k] = s;
    }

    // g = identity (rows 0..2 of 4x4; bottom row implicit [0,0,0,1])
    float g[3][4] = {{1, 0, 0, 0}, {0, 1, 0, 0}, {0, 0, 1, 0}};
    bool done = false;

    for (int it = 0; it < MAXITER_C; ++it) {
        float cj[3];
        #pragma unroll
        for (int j = 0; j < 3; ++j) cj[j] = g[0][j] + g[1][j] + g[2][j];
        const float trs = g[0][3] + g[1][3] + g[2][3];
        const float T = (cj[0] * M[0] + cj[1] * M[1] + cj[2] * M[2]
                         + trs * D1) / 3.0f - DN;
        float dx[6];
        #pragma unroll
        for (int k = 0; k < 6; ++k) dx[k] = -hk[k] * T;

        float n2 = 0.f;
        #pragma unroll
        for (int k = 0; k < 6; ++k) n2 += dx[k] * dx[k];
        float chk = valid ? sqrtf(n2) : 0.0f;

        // wave-wide max, NaN-propagating (to match jnp.max semantics)
        #pragma unroll
        for (int off = 16; off > 0; off >>= 1) {
            float o = __shfl_xor(chk, off, 32);
            float mx = fmaxf(chk, o);
            chk = ((chk != chk) || (o != o)) ? (chk + o) : mx;
        }
        done = done || (chk < XTOL_C);

        if (!done) {
            // se3_exp(dx)
            const float wx = dx[0], wy = dx[1], wz = dx[2];
            const float vx = dx[3], vy = dx[4], vz = dx[5];
            const float t2 = wx * wx + wy * wy + wz * wz;
            const float t  = sqrtf(t2);
            const bool small = t < 1e-4f;
            const float ts = small ? 1.0f : t;
            const float sn = sinf(ts), cs = cosf(ts);
            const float s1 = small ? (1.0f - t2 / 6.0f + t2 * t2 / 120.0f)
                                   : (sn / ts);
            const float s2f = small ? (0.5f - t2 / 24.0f + t2 * t2 / 720.0f)
                                    : ((1.0f - cs) / (ts * ts));
            const float s3 = small ? (1.0f / 6.0f - t2 / 120.0f + t2 * t2 / 5040.0f)
                                   : ((ts - sn) / (ts * ts * ts));
            const float W[3][3] = {{0, -wz, wy}, {wz, 0, -wx}, {-wy, wx, 0}};
            float Smat[3][3];
            #pragma unroll
            for (int i = 0; i < 3; ++i)
                #pragma unroll
                for (int j = 0; j < 3; ++j)
                    Smat[i][j] = W[i][0] * W[0][j] + W[i][1] * W[1][j] + W[i][2] * W[2][j];
            float E[3][4];
            #pragma unroll
            for (int i = 0; i < 3; ++i)
                #pragma unroll
                for (int j = 0; j < 3; ++j) {
                    const float I = (i == j) ? 1.0f : 0.0f;
                    E[i][j] = I + s1 * W[i][j] + s2f * Smat[i][j];   // R
                }
            #pragma unroll
            for (int i = 0; i < 3; ++i) {
                const float Vi0 = (i == 0 ? 1.f : 0.f) + s2f * W[i][0] + s3 * Smat[i][0];
                const float Vi1 = (i == 1 ? 1.f : 0.f) + s2f * W[i][1] + s3 * Smat[i][1];
                const float Vi2 = (i == 2 ? 1.f : 0.f) + s2f * W[i][2] + s3 * Smat[i][2];
                E[i][3] = Vi0 * vx + Vi1 * vy + Vi2 * vz;            // p = V v
            }
            // g = E @ g   (g bottom row = [0,0,0,1])
            float ng[3][4];
            #pragma unroll
            for (int i = 0; i < 3; ++i)
                #pragma unroll
                for (int j = 0; j < 4; ++j) {
                    float s = E[i][0] * g[0][j] + E[i][1] * g[1][j] + E[i][2] * g[2][j];
                    if (j == 3) s += E[i][3];
                    ng[i][j] = s;
                }
            #pragma unroll
            for (int i = 0; i < 3; ++i)
                #pragma unroll
                for (int j = 0; j < 4; ++j) g[i][j] = ng[i][j];
        }
    }

    if (valid) {
        // out = a0 @ [g;0001] @ a1 ;  a0 adds p0m to col 3, a1 subtracts R*p1m
        float* O = out + (size_t)b * 16;
        #pragma unroll
        for (int i = 0; i < 3; ++i) {
            O[i * 4 + 0] = g[i][0];
            O[i * 4 + 1] = g[i][1];
            O[i * 4 + 2] = g[i][2];
            O[i * 4 + 3] = (g[i][3] + p0m[i])
                           - (g[i][0] * p1m[0] + g[i][1] * p1m[1] + g[i][2] * p1m[2]);
        }
        O[12] = 0.f; O[13] = 0.f; O[14] = 0.f; O[15] = 1.f;
    }
}

// ---------------------------------------------------------------------------
extern "C" void kernel_launch(void* const* d_in, const int* in_sizes, int n_in,
                              void* d_out, int out_size, void* d_ws, size_t ws_size,
                              hipStream_t stream) {
    const float* p0  = (const float*)d_in[0];
    const float* p1  = (const float*)d_in[1];
    const float* ndf = (const float*)d_in[2];
    const float* dt  = (const float*)d_in[3];
    float* out = (float*)d_out;

    const int Bn = out_size / 16;            // batches (4x4 outputs)
    const int N  = in_sizes[2] / Bn;         // points per batch (ndf is B*N)

    float* accum = (float*)d_ws;             // Bn * 64 floats

    const int accn = Bn * 64;
    sr_init_kernel<<<(accn + 255) / 256, 256, 0, stream>>>(accum, accn);

    const int numTiles = (N + TP - 1) / TP;
    int chunks = numTiles < 512 ? numTiles : 512;
    dim3 grid(chunks, Bn);
    sr_gram_kernel<<<grid, 256, 0, stream>>>(p0, p1, ndf, accum, N, numTiles);

    sr_finalize_kernel<<<1, 32, 0, stream>>>(accum, dt, out, Bn, N);
}